// ImbalancedLossFunction_14379550507321
// MI455X (gfx1250) — compile-verified
//
#include <hip/hip_runtime.h>

// ---------------------------------------------------------------------------
// Focal-MSE imbalanced loss, MI455X (gfx1250, wave32).
// Memory-bound 2-pass reduction: ~402 MB traffic -> ~17us at 23.3 TB/s.
// Cross-lane reductions use V_WMMA_F32_16X16X4_F32 with a ones B-matrix
// (f32-exact, deterministic), so every kernel exercises the CDNA5 WMMA path.
// ---------------------------------------------------------------------------

#define BLOCKS 1024
#define THREADS 256
#define WAVES_PER_BLOCK (THREADS / 32)

typedef __attribute__((ext_vector_type(2))) float v2f;
typedef __attribute__((ext_vector_type(8))) float v8f;

// Sum of the 32 per-lane values across a wave, computed via WMMA:
//   A[16x4]: VGPR0 = acc (lanes 0..31 cover A[m][0], A[m][2]), VGPR1 = 0
//   B[4x16] = ones  =>  D[m][n] = acc[m] + acc[m+16]  (independent of n)
// Lane n in 0..15 holds D[0..7][n] in its 8 D-VGPRs -> sum = S_low;
// lanes 16..31 hold D[8..15][*]   -> sum = S_high.  Total = S_low + S_high.
__device__ __forceinline__ float wave_sum32(float acc) {
  v2f a; a.x = acc;  a.y = 0.0f;
  v2f b; b.x = 1.0f; b.y = 1.0f;
  v8f c = {};
  c = __builtin_amdgcn_wmma_f32_16x16x4_f32(
      /*neg_a=*/false, a, /*neg_b=*/false, b,
      /*c_mod=*/(short)0, c, /*reuse_a=*/false, /*reuse_b=*/false);
  float s = c[0] + c[1] + c[2] + c[3] + c[4] + c[5] + c[6] + c[7];
  s += __shfl_xor(s, 16, 32);   // combine half-wave sums; all lanes get total
  return s;
}

// Deterministic block reduction: WMMA wave sums -> LDS -> thread 0 adds 8.
__device__ __forceinline__ float block_sum(float acc, float* lds) {
  float w = wave_sum32(acc);
  const int wid  = threadIdx.x >> 5;
  const int lane = threadIdx.x & 31;
  if (lane == 0) lds[wid] = w;
  __syncthreads();
  float r = 0.0f;
  if (threadIdx.x == 0) {
#pragma unroll
    for (int i = 0; i < WAVES_PER_BLOCK; ++i) r += lds[i];
  }
  return r;
}

// Pass 1: per-block partial sums of (p - t)^2. float4 -> global_load_b128.
__global__ void k_mse_partial(const float4* __restrict__ p,
                              const float4* __restrict__ t,
                              long long n4, int nrem,
                              const float* __restrict__ prem,
                              const float* __restrict__ trem,
                              float* __restrict__ partials) {
  __shared__ float lds[WAVES_PER_BLOCK];
  long long idx    = (long long)blockIdx.x * blockDim.x + threadIdx.x;
  long long stride = (long long)gridDim.x * blockDim.x;
  float acc = 0.0f;
  for (long long i = idx; i < n4; i += stride) {
    const float4 a = p[i];
    const float4 b = t[i];
    const float d0 = a.x - b.x, d1 = a.y - b.y, d2 = a.z - b.z, d3 = a.w - b.w;
    acc = fmaf(d0, d0, acc);
    acc = fmaf(d1, d1, acc);
    acc = fmaf(d2, d2, acc);
    acc = fmaf(d3, d3, acc);
  }
  if (idx == 0) {                      // scalar tail (empty for this shape)
    for (int r = 0; r < nrem; ++r) {
      const float d = prem[r] - trem[r];
      acc = fmaf(d, d, acc);
    }
  }
  const float bs = block_sum(acc, lds);  // reconverged: EXEC all 1s for WMMA
  if (threadIdx.x == 0) partials[blockIdx.x] = bs;
}

// Reduce the 1024 partials -> s = mean + 1e-8 (double for the final divide).
__global__ void k_finalize_msum(const float* __restrict__ partials,
                                float* __restrict__ s_out, double inv_n) {
  __shared__ float lds[WAVES_PER_BLOCK];
  float acc = 0.0f;
  for (int i = threadIdx.x; i < BLOCKS; i += THREADS) acc += partials[i];
  const float tot = block_sum(acc, lds);
  if (threadIdx.x == 0) {
    const double mean = (double)tot * inv_n;
    s_out[0] = (float)(mean + 1e-8);
  }
}

// Pass 2: per-block partial sums of (m/(s+m))^2 * m,  m = (p-t)^2.
__global__ void k_focal_partial(const float4* __restrict__ p,
                                const float4* __restrict__ t,
                                long long n4, int nrem,
                                const float* __restrict__ prem,
                                const float* __restrict__ trem,
                                const float* __restrict__ s_ptr,
                                float* __restrict__ partials) {
  __shared__ float lds[WAVES_PER_BLOCK];
  const float s = s_ptr[0];            // uniform broadcast load
  long long idx    = (long long)blockIdx.x * blockDim.x + threadIdx.x;
  long long stride = (long long)gridDim.x * blockDim.x;
  float acc = 0.0f;
  for (long long i = idx; i < n4; i += stride) {
    const float4 a = p[i];
    const float4 b = t[i];
    const float d[4] = {a.x - b.x, a.y - b.y, a.z - b.z, a.w - b.w};
#pragma unroll
    for (int j = 0; j < 4; ++j) {
      const float m = d[j] * d[j];
      const float r = m / (s + m);     // = 1 - confidence
      acc = fmaf(r * r, m, acc);
    }
  }
  if (idx == 0) {
    for (int k = 0; k < nrem; ++k) {
      const float dd = prem[k] - trem[k];
      const float m  = dd * dd;
      const float r  = m / (s + m);
      acc = fmaf(r * r, m, acc);
    }
  }
  const float bs = block_sum(acc, lds);
  if (threadIdx.x == 0) partials[blockIdx.x] = bs;
}

// Final: focal = 0.25 * fsum / N; out = focal * mean(DW[ids]).
__global__ void k_final(const float* __restrict__ fpartials,
                        const int* __restrict__ ids, int batch,
                        double inv_n, float* __restrict__ out) {
  __shared__ float lds[WAVES_PER_BLOCK];
  float acc = 0.0f;
  for (int i = threadIdx.x; i < BLOCKS; i += THREADS) acc += fpartials[i];
  const float tot = block_sum(acc, lds);
  if (threadIdx.x == 0) {
    const float DW[3] = {0.6502451783856802f, 1.449137674618944f,
                         2.509980079602226f};
    double wsum = 0.0;
    for (int b = 0; b < batch; ++b) {
      int id = ids[b];
      id = id < 0 ? 0 : (id > 2 ? 2 : id);
      wsum += (double)DW[id];
    }
    const double focal = 0.25 * (double)tot * inv_n;
    out[0] = (float)(focal * (wsum / (double)batch));
  }
}

extern "C" void kernel_launch(void* const* d_in, const int* in_sizes, int n_in,
                              void* d_out, int out_size, void* d_ws,
                              size_t ws_size, hipStream_t stream) {
  const float* pred = (const float*)d_in[0];
  const float* targ = (const float*)d_in[1];
  const int*   ids  = (const int*)d_in[2];
  const int    batch = in_sizes[2];

  const long long n    = (long long)in_sizes[0];
  const long long n4   = n >> 2;
  const int       nrem = (int)(n - (n4 << 2));
  const double    inv_n = 1.0 / (double)n;

  // Workspace layout (floats): [0..1023] mse partials, [1024] s,
  //                            [1025..2048] focal partials.  (8.2 KB)
  float* ws        = (float*)d_ws;
  float* mse_part  = ws;
  float* s_scalar  = ws + BLOCKS;
  float* foc_part  = ws + BLOCKS + 1;

  const float* ptail = pred + (n4 << 2);
  const float* ttail = targ + (n4 << 2);

  k_mse_partial<<<BLOCKS, THREADS, 0, stream>>>(
      (const float4*)pred, (const float4*)targ, n4, nrem, ptail, ttail,
      mse_part);
  k_finalize_msum<<<1, THREADS, 0, stream>>>(mse_part, s_scalar, inv_n);
  k_focal_partial<<<BLOCKS, THREADS, 0, stream>>>(
      (const float4*)pred, (const float4*)targ, n4, nrem, ptail, ttail,
      s_scalar, foc_part);
  k_final<<<1, THREADS, 0, stream>>>(foc_part, ids, batch, inv_n,
                                     (float*)d_out);
}